// JODIE_80255758893186
// MI455X (gfx1250) — compile-verified
//
#include <hip/hip_runtime.h>

// ---------------- problem constants ----------------
#define NUSERS   4000
#define NLOCAS   4000
#define DD       128
#define KK       64
#define TT       2048
#define PREDIN   8320            // NUSERS + NLOCAS + 2*DD + KK
#define PREDOUT  4192            // NLOCAS + DD + KK
#define INRNN    193             // DD + 1 + KK
#define NTILES   262             // PREDOUT / 16
#define NKB      80              // 320 / 4 (K-blocks of the f32 WMMA)
#define NBLK     67              // 66 GEMV blocks + 1 RNN block
#define RNNBLK   (NBLK - 1)
#define BTHREADS 128             // 4 waves (wave32)
#define EMB_N    (8000 * DD)     // 1,024,000
#define PACKED_FLOATS ((size_t)NTILES * NKB * 64)   // 1,341,440
#define WS_PACKED_OFF 4096
#define WS_NEED (WS_PACKED_OFF + PACKED_FLOATS * sizeof(float))

typedef float v2f __attribute__((ext_vector_type(2)));
typedef float v8f __attribute__((ext_vector_type(8)));

// ---------------- device-wide barrier (persistent-kernel sync) ----------------
__device__ __forceinline__ void grid_sync(unsigned* cnt, unsigned* gen) {
  __syncthreads();
  if (threadIdx.x == 0) {
    __threadfence();  // release: publish prior global writes
    unsigned g0  = __hip_atomic_load(gen, __ATOMIC_RELAXED, __HIP_MEMORY_SCOPE_AGENT);
    unsigned arr = __hip_atomic_fetch_add(cnt, 1u, __ATOMIC_ACQ_REL, __HIP_MEMORY_SCOPE_AGENT);
    if (arr == (unsigned)(NBLK - 1)) {
      __hip_atomic_store(cnt, 0u, __ATOMIC_RELAXED, __HIP_MEMORY_SCOPE_AGENT);
      __hip_atomic_fetch_add(gen, 1u, __ATOMIC_RELEASE, __HIP_MEMORY_SCOPE_AGENT);
    } else {
      while (__hip_atomic_load(gen, __ATOMIC_ACQUIRE, __HIP_MEMORY_SCOPE_AGENT) == g0) {
        __builtin_amdgcn_s_sleep(2);
      }
    }
    __threadfence();  // acquire
  }
  __syncthreads();
}

// ---------------- helper kernels ----------------
__global__ void k_init(unsigned* cnt, unsigned* gen) {
  *cnt = 0u; *gen = 0u;
}

__global__ void k_copy(const float* __restrict__ src, float* __restrict__ dst, int n) {
  int i = blockIdx.x * blockDim.x + threadIdx.x;
  if (i < n) dst[i] = src[i];
}

// Pack A-tiles of the dense 4192x320 weight block into WMMA f32 16x16x4 lane
// order: lane l<16 holds W[tile*16+l][kb*4+{0,1}], lane l>=16 holds K={2,3}.
__global__ void k_pack(const float* __restrict__ pred_w, float* __restrict__ packed) {
  int g = blockIdx.x * blockDim.x + threadIdx.x;
  if (g >= NTILES * NKB * 32) return;
  int l    = g & 31;
  int rest = g >> 5;
  int kb   = rest % NKB;
  int tile = rest / NKB;
  int row  = tile * 16 + (l & 15);
  int c    = kb * 4 + ((l >> 4) << 1);
  size_t src = (size_t)row * PREDIN + c;
  packed[2 * (size_t)g]     = pred_w[src];
  packed[2 * (size_t)g + 1] = pred_w[src + 1];
}

__global__ void k_final(const float* __restrict__ partials, float* __restrict__ loss) {
  if (threadIdx.x == 0) {
    float s = 0.f;
    for (int i = 0; i < NBLK; ++i) s += partials[i];   // deterministic order
    *loss = s;
  }
}

// ---------------- main persistent kernel ----------------
__global__ void __launch_bounds__(BTHREADS, 1)
jodie_main(float* __restrict__ emb,                 // working copy in d_out
           const int*   __restrict__ ev,
           const float* __restrict__ dus,
           const float* __restrict__ dls,
           const float* __restrict__ kg,
           const float* __restrict__ w_ih_u, const float* __restrict__ w_hh_u,
           const float* __restrict__ b_ih_u, const float* __restrict__ b_hh_u,
           const float* __restrict__ w_ih_l, const float* __restrict__ w_hh_l,
           const float* __restrict__ b_ih_l, const float* __restrict__ b_hh_l,
           const float* __restrict__ pw, const float* __restrict__ pb,
           const float* __restrict__ pred_w, const float* __restrict__ pred_b,
           const float* __restrict__ packedA, int usePacked,
           unsigned* cnt, unsigned* gen, float* __restrict__ partials) {
  // smem: GEMV block: x[0..319], e_l[320..447], final reduce[0..127] (reuses x)
  //       RNN  block: e_u[0..127], e_l[128..255], kg_n[256..319], red_u[320..447], red_l[448..575]
  __shared__ float smem[576];
  const int tid  = threadIdx.x;
  const int bid  = blockIdx.x;
  const int lane = tid & 31;
  const int wave = tid >> 5;
  const bool is_rnn = (bid == RNNBLK);
  const int tile = bid * 4 + wave;           // valid for GEMV blocks, 0..263
  const int sel  = lane >> 4;                // 0: K pair {0,1} / rows 0..7 ; 1: {2,3} / rows 8..15

  // Loop-invariant A base pointer for this wave (packed path): base + kb*32 v2f
  const v2f* ap_base = (const v2f*)packedA + (size_t)tile * NKB * 32 + lane;
  // Fallback (unpacked) base: row-major pred_w
  const float* aw_base = pred_w + (size_t)(tile * 16 + (lane & 15)) * PREDIN + sel * 2;

  float lossAcc = 0.f;
  float r_newu = 0.f, r_newl = 0.f;
  int   r_u = 0, r_lg = 0;

  for (int t = 0; t < TT; ++t) {
    grid_sync(cnt, gen);                     // publishes step t-1 emb writes

    const int u    = ev[t * 5 + 0];
    const int loca = ev[t * 5 + 1];
    const int prev = ev[t * 5 + 2];
    const int kp   = ev[t * 5 + 3];
    const int kn   = ev[t * 5 + 4];
    const float du = dus[t];
    const float dl = dls[t];
    const int lg = loca + NUSERS;
    const int pg = prev + NUSERS;

    if (!is_rnn) {
      // ---- stage x = [proj ; e_prev ; kg[know_prev]] and e_l snapshot into LDS ----
      {
        float e_u_i = emb[u * DD + tid];
        smem[tid]       = e_u_i * (1.0f + pw[tid] * du + pb[tid]);   // proj
        smem[128 + tid] = emb[pg * DD + tid];                        // e_prev
        smem[320 + tid] = emb[lg * DD + tid];                        // e_l (target rows < 128)
        if (tid < KK) smem[256 + tid] = kg[kp * KK + tid];           // kg[know_prev]
      }
      __syncthreads();

      if (tile < NTILES) {
        v8f acc = {0.f, 0.f, 0.f, 0.f, 0.f, 0.f, 0.f, 0.f};
        if (usePacked) {
          // hot path: single base pointer + constant byte offsets -> clauseable loads
          #pragma unroll 8
          for (int kb = 0; kb < NKB; ++kb) {
            v2f a = ap_base[kb * 32];
            v2f b;
            b.x = smem[kb * 4 + sel];
            b.y = smem[kb * 4 + 2 + sel];
            // D(16x16,f32) = A(16x4,f32) x B(4x16,f32) + C   [exact fp32 path]
            acc = __builtin_amdgcn_wmma_f32_16x16x4_f32(false, a, false, b,
                                                        (short)0, acc, false, false);
          }
        } else {
          #pragma unroll 4
          for (int kb = 0; kb < NKB; ++kb) {
            v2f a;
            a.x = aw_base[kb * 4];
            a.y = aw_base[kb * 4 + 1];
            v2f b;
            b.x = smem[kb * 4 + sel];
            b.y = smem[kb * 4 + 2 + sel];
            acc = __builtin_amdgcn_wmma_f32_16x16x4_f32(false, a, false, b,
                                                        (short)0, acc, false, false);
          }
        }
        // epilogue: all 16 N-columns of D are identical (B broadcast) -> 1/16 scale
        const size_t csl = 320 + (size_t)prev;    // W_sl column
        const size_t csu = 4320 + (size_t)u;      // W_su column
        const int rbase = tile * 16 + 8 * sel;
        #pragma unroll
        for (int v = 0; v < 8; ++v) {
          int row = rbase + v;
          float val = acc[v]
                    + pred_w[(size_t)row * PREDIN + csl]
                    + pred_w[(size_t)row * PREDIN + csu]
                    + pred_b[row];
          float tgt;
          if (row < DD)                tgt = smem[320 + row];
          else if (row < DD + NLOCAS)  tgt = (row - DD == prev) ? 1.0f : 0.0f;
          else                         tgt = kg[kn * KK + (row - DD - NLOCAS)];
          float d = val - tgt;
          lossAcc += d * d;
        }
      }
    } else {
      // ---- RNN cell updates (128 threads, one row each) ----
      const int r = tid;
      smem[r]       = emb[u * DD + r];     // e_u (old)
      smem[128 + r] = emb[lg * DD + r];    // e_l (old)
      if (r < KK) smem[256 + r] = kg[kn * KK + r];
      __syncthreads();

      float au = b_ih_u[r] + b_hh_u[r];
      const float* wiu = w_ih_u + r * INRNN;
      for (int c = 0; c < DD; ++c) au += wiu[c] * smem[128 + c];        // e_l part
      for (int c = 0; c < KK; ++c) au += wiu[DD + c] * smem[256 + c];   // kg part
      au += wiu[192] * du;
      const float* whu = w_hh_u + r * DD;
      for (int c = 0; c < DD; ++c) au += whu[c] * smem[c];              // e_u part
      float tu = tanhf(au);

      float al = b_ih_l[r] + b_hh_l[r];
      const float* wil = w_ih_l + r * INRNN;
      for (int c = 0; c < DD; ++c) al += wil[c] * smem[c];              // e_u part
      for (int c = 0; c < KK; ++c) al += wil[DD + c] * smem[256 + c];   // kg part
      al += wil[192] * dl;
      const float* whl = w_hh_l + r * DD;
      for (int c = 0; c < DD; ++c) al += whl[c] * smem[128 + c];        // e_l part
      float tl = tanhf(al);

      smem[320 + r] = tu * tu;
      smem[448 + r] = tl * tl;
      __syncthreads();
      for (int s = 64; s > 0; s >>= 1) {
        if (r < s) { smem[320 + r] += smem[320 + r + s]; smem[448 + r] += smem[448 + r + s]; }
        __syncthreads();
      }
      float nu = fmaxf(sqrtf(smem[320]), 1e-12f);
      float nl = fmaxf(sqrtf(smem[448]), 1e-12f);
      float newu = tu / nu;
      float newl = tl / nl;
      float d0 = newu - smem[r];
      float d1 = newl - smem[128 + r];
      lossAcc += d0 * d0 + d1 * d1;
      r_newu = newu; r_newl = newl; r_u = u; r_lg = lg;
    }

    grid_sync(cnt, gen);                     // all reads of emb for step t done
    if (is_rnn) {                            // commit embedding updates
      emb[r_u * DD + tid]  = r_newu;
      emb[r_lg * DD + tid] = r_newl;
    }
  }

  // ---- deterministic per-block loss reduction ----
  float scale = is_rnn ? (1.0f / 128.0f) : (1.0f / (16.0f * 4192.0f));
  __syncthreads();
  smem[tid] = lossAcc * scale;
  __syncthreads();
  for (int s = 64; s > 0; s >>= 1) {
    if (tid < s) smem[tid] += smem[tid + s];
    __syncthreads();
  }
  if (tid == 0) partials[bid] = smem[0];
}

// ---------------- host entry ----------------
extern "C" void kernel_launch(void* const* d_in, const int* in_sizes, int n_in,
                              void* d_out, int out_size, void* d_ws, size_t ws_size,
                              hipStream_t stream) {
  const float* embedding = (const float*)d_in[0];
  const int*   ev        = (const int*)  d_in[1];
  const float* dus       = (const float*)d_in[2];
  const float* dls       = (const float*)d_in[3];
  const float* kg        = (const float*)d_in[4];
  const float* w_ih_u    = (const float*)d_in[5];
  const float* w_hh_u    = (const float*)d_in[6];
  const float* b_ih_u    = (const float*)d_in[7];
  const float* b_hh_u    = (const float*)d_in[8];
  const float* w_ih_l    = (const float*)d_in[9];
  const float* w_hh_l    = (const float*)d_in[10];
  const float* b_ih_l    = (const float*)d_in[11];
  const float* b_hh_l    = (const float*)d_in[12];
  const float* proj_w    = (const float*)d_in[13];
  const float* proj_b    = (const float*)d_in[14];
  const float* pred_w    = (const float*)d_in[15];
  const float* pred_b    = (const float*)d_in[16];

  float* out  = (float*)d_out;
  float* emb  = out;                 // 8000*128 floats, final state = output
  float* loss = out + EMB_N;

  char*     ws       = (char*)d_ws;
  unsigned* cnt      = (unsigned*)(ws + 0);
  unsigned* gen      = (unsigned*)(ws + 4);
  float*    partials = (float*)(ws + 256);
  float*    packed   = (float*)(ws + WS_PACKED_OFF);
  int usePacked = (ws_size >= WS_NEED) ? 1 : 0;

  k_init<<<1, 1, 0, stream>>>(cnt, gen);
  k_copy<<<(EMB_N + 511) / 512, 512, 0, stream>>>(embedding, emb, EMB_N);
  if (usePacked) {
    k_pack<<<(NTILES * NKB * 32 + 255) / 256, 256, 0, stream>>>(pred_w, packed);
  }
  jodie_main<<<NBLK, BTHREADS, 0, stream>>>(
      emb, ev, dus, dls, kg,
      w_ih_u, w_hh_u, b_ih_u, b_hh_u,
      w_ih_l, w_hh_l, b_ih_l, b_hh_l,
      proj_w, proj_b, pred_w, pred_b,
      packed, usePacked, cnt, gen, partials);
  k_final<<<1, 1, 0, stream>>>(partials, loss);
}